// PolicyLP_PlusNeq_66477503808090
// MI455X (gfx1250) — compile-verified
//
#include <hip/hip_runtime.h>
#include <hip/hip_bf16.h>

// CDNA5 (gfx1250) batched interior-point QP solver.
// One workgroup per batch element; all state in LDS; structure-exploited
// Schur complement formed with V_WMMA_F32_16X16X4_F32 on wave 0; the whole
// 65x65 block-solve runs inside wave 0 with wave-local ordering (wave32).

typedef __attribute__((ext_vector_type(2))) float v2f;
typedef __attribute__((ext_vector_type(8))) float v8f;

#define NSIDE    32
#define NV       1024      // n = 32*32
#define QP_EPS   0.1f
#define QP_SIGMA 0.1f
#define N_ITERS  30
#define NTHREADS 256

// Wave-local ordering point: scheduler fence + LDS register-caching fence.
// Cross-lane LDS deps within one wave are in-order on CDNA5; this stops the
// compiler from reordering or caching LDS values across the point.
__device__ __forceinline__ void wave_fence() {
    __builtin_amdgcn_wave_barrier();
    asm volatile("" ::: "memory");
}

__device__ __forceinline__ float block_sum(float v, float* red, int tid) {
    red[tid] = v; __syncthreads();
    #pragma unroll
    for (int off = NTHREADS / 2; off > 0; off >>= 1) {
        if (tid < off) red[tid] += red[tid + off];
        __syncthreads();
    }
    float r = red[0]; __syncthreads();
    return r;
}

__device__ __forceinline__ float block_min(float v, float* red, int tid) {
    red[tid] = v; __syncthreads();
    #pragma unroll
    for (int off = NTHREADS / 2; off > 0; off >>= 1) {
        if (tid < off) red[tid] = fminf(red[tid], red[tid + off]);
        __syncthreads();
    }
    float r = red[0]; __syncthreads();
    return r;
}

__global__ __launch_bounds__(NTHREADS)
void policy_qp_ipm_kernel(const float* __restrict__ x,
                          const float* __restrict__ g,
                          const float* __restrict__ expo,
                          float* __restrict__ out) {
    const int b   = blockIdx.x;
    const int tid = threadIdx.x;

    // ---- LDS state (~49 KB/WG; 320 KB/WGP budget) ----
    __shared__ float z[NV], p[NV], vxf[NV];
    __shared__ float s_lo[NV], s_hi[NV], l_lo[NV], l_hi[NV];
    __shared__ float Hinv[NV], rhs1[NV], dzs[NV];
    __shared__ float yv[72], dyv[72], rp[72], rhsy[72];
    __shared__ float dr[32], dcc[32], isd[32], uu[32], vv[32];
    __shared__ float Gm[32 * 33];                 // stride 33: bank-conflict pad
    __shared__ float c2[32], rr[32], r0[32], qv[32], tv[32];
    __shared__ float farr[32], exv[32], gsh[32];
    __shared__ float red[NTHREADS];
    __shared__ float scal[8];

    // gfx1250 global_prefetch_b8 on this block's x row
    __builtin_prefetch((const void*)(x + (size_t)b * NV), 0, 1);

    // ---- init: f = g/sum(g) - (1-g)/sum(1-g), vXf = f ⊗ exposure ----
    if (tid < 32) { exv[tid] = expo[tid]; gsh[tid] = g[b * 32 + tid]; }
    if (tid < 72) { yv[tid] = 0.0f; }
    __syncthreads();
    if (tid < 32) {
        float sg = 0.0f;
        #pragma unroll
        for (int i = 0; i < 32; ++i) sg += gsh[i];
        float gi = gsh[tid];
        farr[tid] = gi / sg - (1.0f - gi) / (32.0f - sg);
    }
    __syncthreads();
    for (int k = tid; k < NV; k += NTHREADS) {
        int i = k >> 5, j = k & 31;
        p[k]    = -x[(size_t)b * NV + k];
        vxf[k]  = farr[i] * exv[j];
        z[k]    = 0.5f;
        s_lo[k] = 0.5f; s_hi[k] = 0.5f;
        l_lo[k] = 1.0f; l_hi[k] = 1.0f;
    }
    __syncthreads();

    for (int it = 0; it < N_ITERS; ++it) {
        // ---- mu = mean(lam*s) over 2n ----
        float lm = 0.0f;
        for (int k = tid; k < NV; k += NTHREADS)
            lm += l_lo[k] * s_lo[k] + l_hi[k] * s_hi[k];
        float mu = block_sum(lm, red, tid) * (1.0f / 2048.0f);
        if (tid == 0) scal[0] = mu;

        // ---- elementwise: Hinv, rhs1 (Aᵀy via structure) ----
        for (int k = tid; k < NV; k += NTHREADS) {
            int i = k >> 5, j = k & 31;
            float zk = z[k], slo = s_lo[k], shi = s_hi[k];
            float llo = l_lo[k], lhi = l_hi[k];
            float aty  = yv[i] + yv[32 + j] + yv[64] * vxf[k];
            float rd   = QP_EPS * zk + p[k] + aty + (lhi - llo);
            float rglo = -zk + slo;
            float rghi =  zk + shi - 1.0f;
            float wlo  = llo / slo, whi = lhi / shi;
            float hv   = 1.0f / (QP_EPS + wlo + whi);
            float rclo = llo * slo - QP_SIGMA * mu;
            float rchi = lhi * shi - QP_SIGMA * mu;
            float ulo  = wlo * rglo - rclo / slo;
            float uhi  = whi * rghi - rchi / shi;
            Hinv[k] = hv;
            rhs1[k] = -rd + ulo - uhi;
        }
        __syncthreads();

        // ---- r_p rows/cols, dr/dc/u/v, rhs_y rows/cols (A's sparsity) ----
        if (tid < 32) {
            int t = tid;
            float rsz = 0.f, csz = 0.f, drr = 0.f, dcv = 0.f;
            float uval = 0.f, vval = 0.f, ry = 0.f, cy = 0.f;
            #pragma unroll 4
            for (int j2 = 0; j2 < 32; ++j2) {
                int kr = t * 32 + j2;   // row t of the 32x32 grid
                int kc = j2 * 32 + t;   // col t
                float hr = Hinv[kr], hc = Hinv[kc];
                rsz += z[kr]; csz += z[kc];
                drr += hr;    dcv += hc;
                uval += hr * vxf[kr]; vval += hc * vxf[kc];
                ry   += hr * rhs1[kr]; cy  += hc * rhs1[kc];
            }
            rp[t]      = rsz - 1.0f;
            rp[32 + t] = csz - 1.0f;
            dr[t] = drr; dcc[t] = dcv; uu[t] = uval; vv[t] = vval;
            isd[t] = rsqrtf(drr);
            rhsy[t]      = ry + rp[t];
            rhsy[32 + t] = cy + rp[32 + t];
        }
        // ---- scalar reductions: rp[64], w, rhsy[64] ----
        float a0 = 0.f, a1 = 0.f, a2 = 0.f;
        for (int k = tid; k < NV; k += NTHREADS) {
            float vx = vxf[k], hv = Hinv[k];
            a0 += vx * z[k];
            a1 += hv * vx * vx;
            a2 += vx * hv * rhs1[k];
        }
        float rp64  = block_sum(a0, red, tid);
        float wfull = block_sum(a1, red, tid);
        float ry64p = block_sum(a2, red, tid);
        if (tid == 0) { rp[64] = rp64; rhsy[64] = ry64p + rp64; scal[1] = wfull; }
        __syncthreads();

        // ================= wave-0-only block solve =================
        if (tid < 32) {
            const int l15  = tid & 15;
            const int koff = (tid < 16) ? 0 : 2;   // A/B f32 K-layout split
            const int roff = (tid < 16) ? 0 : 8;   // C/D f32 M-layout split

            // -- G = Dc - Mhᵀ·Mh via V_WMMA_F32_16X16X4_F32;
            //    Mh[i][j] = Hinv[i*32+j] * isd[i] --
            #pragma unroll
            for (int Mi = 0; Mi <= 16; Mi += 16) {
                #pragma unroll
                for (int Nj = 0; Nj <= 16; Nj += 16) {
                    v8f acc = {};
                    #pragma unroll
                    for (int kb = 0; kb < 32; kb += 4) {
                        int k0 = kb + koff, k1 = kb + koff + 1;
                        v2f av, bv;
                        av.x = Hinv[k0 * 32 + Mi + l15] * isd[k0];
                        av.y = Hinv[k1 * 32 + Mi + l15] * isd[k1];
                        bv.x = Hinv[k0 * 32 + Nj + l15] * isd[k0];
                        bv.y = Hinv[k1 * 32 + Nj + l15] * isd[k1];
                        acc = __builtin_amdgcn_wmma_f32_16x16x4_f32(
                            /*neg_a=*/false, av, /*neg_b=*/false, bv,
                            /*c_mod=*/(short)0, acc,
                            /*reuse_a=*/false, /*reuse_b=*/false);
                    }
                    #pragma unroll
                    for (int v = 0; v < 8; ++v)
                        Gm[(Mi + roff + v) * 33 + Nj + l15] = -acc[v];
                }
            }
            wave_fence();
            // divergence-free diagonal add
            Gm[tid * 33 + tid] += dcc[tid];

            // -- c2 = b2 - MᵀDr⁻¹b1 ; r = v - MᵀDr⁻¹u --
            {
                float cc = 0.f, rv = 0.f;
                #pragma unroll 4
                for (int i2 = 0; i2 < 32; ++i2) {
                    float mi = Hinv[i2 * 32 + tid] / dr[i2];
                    cc += mi * rhsy[i2];
                    rv += mi * uu[i2];
                }
                c2[tid] = rhsy[32 + tid] - cc;
                float rt = vv[tid] - rv;
                rr[tid] = rt; r0[tid] = rt;
            }
            if (tid == 0) {
                float w2 = scal[1], b32 = rhsy[64];
                for (int i2 = 0; i2 < 32; ++i2) {
                    float ui = uu[i2] / dr[i2];
                    w2  -= ui * uu[i2];
                    b32 -= ui * rhsy[i2];
                }
                scal[2] = w2; scal[3] = b32;
            }
            wave_fence();

            // -- Gauss-Jordan on [G | c2 | rr]; lane t owns row t.
            //    Pivot lane writes its row back unchanged (f==0), so the
            //    concurrent broadcast reads of the pivot row are race-free. --
            for (int kk = 0; kk < 32; ++kk) {
                float piv  = Gm[kk * 33 + kk];
                float c2k  = c2[kk];
                float rrk  = rr[kk];
                float f    = (tid != kk) ? (Gm[tid * 33 + kk] / piv) : 0.0f;
                for (int c = kk; c < 32; ++c) {
                    float gkc = Gm[kk * 33 + c];     // broadcast read
                    Gm[tid * 33 + c] -= f * gkc;
                }
                c2[tid] -= f * c2k;
                rr[tid] -= f * rrk;
                wave_fence();
            }
            {
                float dinv = 1.0f / Gm[tid * 33 + tid];
                qv[tid] = c2[tid] * dinv;
                tv[tid] = rr[tid] * dinv;
            }
            wave_fence();
            if (tid == 0) {
                float rq = 0.f, rt = 0.f;
                for (int j2 = 0; j2 < 32; ++j2) {
                    rq += r0[j2] * qv[j2];
                    rt += r0[j2] * tv[j2];
                }
                float y3 = (scal[3] - rq) / (scal[2] - rt);
                dyv[64] = y3; scal[4] = y3;
            }
            wave_fence();
            dyv[32 + tid] = qv[tid] - scal[4] * tv[tid];
            wave_fence();
            {
                float s2 = 0.f;
                #pragma unroll 4
                for (int j2 = 0; j2 < 32; ++j2)
                    s2 += Hinv[tid * 32 + j2] * dyv[32 + j2];
                dyv[tid] = (rhsy[tid] - s2 - uu[tid] * scal[4]) / dr[tid];
            }
        }
        __syncthreads();
        // ================ end wave-0-only block solve ================

        // ---- dz, step-length (fraction-to-boundary) ----
        float amin = 1e10f;
        float mu2  = scal[0];
        for (int k = tid; k < NV; k += NTHREADS) {
            int i = k >> 5, j = k & 31;
            float ad  = dyv[i] + dyv[32 + j] + dyv[64] * vxf[k];
            float dzk = Hinv[k] * (rhs1[k] - ad);
            dzs[k] = dzk;
            float zk = z[k], slo = s_lo[k], shi = s_hi[k];
            float llo = l_lo[k], lhi = l_hi[k];
            float rglo = -zk + slo, rghi = zk + shi - 1.0f;
            float wlo = llo / slo, whi = lhi / shi;
            float rclo = llo * slo - QP_SIGMA * mu2;
            float rchi = lhi * shi - QP_SIGMA * mu2;
            float dslo = -rglo + dzk, dshi = -rghi - dzk;
            float dllo = wlo * (-dzk + rglo) - rclo / slo;
            float dlhi = whi * ( dzk + rghi) - rchi / shi;
            if (dslo < 0.f) amin = fminf(amin, -slo / dslo);
            if (dshi < 0.f) amin = fminf(amin, -shi / dshi);
            if (dllo < 0.f) amin = fminf(amin, -llo / dllo);
            if (dlhi < 0.f) amin = fminf(amin, -lhi / dlhi);
        }
        float astep = block_min(amin, red, tid);
        float alpha = fminf(1.0f, 0.99f * astep);

        // ---- updates ----
        for (int k = tid; k < NV; k += NTHREADS) {
            float zk = z[k], slo = s_lo[k], shi = s_hi[k];
            float llo = l_lo[k], lhi = l_hi[k];
            float dzk  = dzs[k];
            float rglo = -zk + slo, rghi = zk + shi - 1.0f;
            float wlo = llo / slo, whi = lhi / shi;
            float rclo = llo * slo - QP_SIGMA * mu2;
            float rchi = lhi * shi - QP_SIGMA * mu2;
            float dslo = -rglo + dzk, dshi = -rghi - dzk;
            float dllo = wlo * (-dzk + rglo) - rclo / slo;
            float dlhi = whi * ( dzk + rghi) - rchi / shi;
            z[k]    = zk  + alpha * dzk;
            s_lo[k] = slo + alpha * dslo;
            s_hi[k] = shi + alpha * dshi;
            l_lo[k] = llo + alpha * dllo;
            l_hi[k] = lhi + alpha * dlhi;
        }
        if (tid < 32) { yv[tid] += alpha * dyv[tid]; yv[32 + tid] += alpha * dyv[32 + tid]; }
        if (tid == 0) { yv[64] += alpha * dyv[64]; }
        __syncthreads();
    }

    // ---- write z ----
    for (int k = tid; k < NV; k += NTHREADS)
        out[(size_t)b * NV + k] = z[k];
}

extern "C" void kernel_launch(void* const* d_in, const int* in_sizes, int n_in,
                              void* d_out, int out_size, void* d_ws, size_t ws_size,
                              hipStream_t stream) {
    (void)n_in; (void)out_size; (void)d_ws; (void)ws_size;
    const float* x    = (const float*)d_in[0];   // (B, 1024) f32
    const float* g    = (const float*)d_in[1];   // (B, 32)   f32
    const float* expo = (const float*)d_in[2];   // (32,)     f32
    float* out = (float*)d_out;                  // (B, 32, 32) f32
    const int batch = in_sizes[0] / NV;          // 128
    policy_qp_ipm_kernel<<<batch, NTHREADS, 0, stream>>>(x, g, expo, out);
}